// TransformerNet_7310034338503
// MI455X (gfx1250) — compile-verified
//
#include <hip/hip_runtime.h>
#include <math.h>

// ---------------------------------------------------------------------------
// TransformerNet on MI455X (gfx1250): fp32 flash-attention via
// V_WMMA_F32_16X16X4_F32 computed TRANSPOSED (S^T = K Q^T, O^T = V^T P^T)
// so softmax rows and the P operand are lane-local: no LDS, no barriers,
// 2 shuffles per tile. Q is pre-scaled by 1/sqrt(D) before the key loop.
// Activations are L2-resident; scores never hit memory.
// ---------------------------------------------------------------------------

typedef float v2f __attribute__((ext_vector_type(2)));
typedef float v8f __attribute__((ext_vector_type(8)));

constexpr int kB = 8, kS = 2048, kE = 32, kH = 4, kD = 8, kFF = 128, kL = 6;
constexpr float kEps = 1e-5f;

static __device__ __forceinline__ v8f wmma_f32_16x16x4(v2f a, v2f b, v8f c) {
  // (neg_a, A, neg_b, B, c_mod, C, reuse_a, reuse_b)
  return __builtin_amdgcn_wmma_f32_16x16x4_f32(false, a, false, b, (short)0, c,
                                               false, false);
}

// ---------------------------------------------------------------------------
// x[b,s,e] = src[b,s,e] + pe[b,e]   (reference quirk: PE indexed by batch)
// ---------------------------------------------------------------------------
__global__ void posadd_kernel(const float* __restrict__ src,
                              float* __restrict__ x) {
  int idx = blockIdx.x * blockDim.x + threadIdx.x;  // over B*S*E = 524288
  int e = idx & 31;
  int b = idx >> 16;  // S*E = 65536
  int i = e >> 1;
  float dv = __expf(-(float)(2 * i) * 0.28782313662425574f);  // ln(1e4)/32
  float ang = (float)b * dv;
  float pe = (e & 1) ? cosf(ang) : sinf(ang);
  x[idx] = src[idx] + pe;
}

// ---------------------------------------------------------------------------
// q[b,h,s,e] = sum_d x[b,s,h*8+d] * Wq[e,d]   (same for k,v); layout [b][h][s][d]
// ---------------------------------------------------------------------------
__global__ void qkv_kernel(const float* __restrict__ x,
                           const float* __restrict__ wq,
                           const float* __restrict__ wk,
                           const float* __restrict__ wv,
                           float* __restrict__ q, float* __restrict__ k,
                           float* __restrict__ v) {
  int t = blockIdx.x * blockDim.x + threadIdx.x;  // over B*H*S = 65536
  int rem = t & (kH * kS - 1);
  int b = t >> 13;
  int h = rem >> 11;
  int s = rem & (kS - 1);
  const float* xp = x + ((size_t)b * kS + s) * kE + h * kD;
  float xr[8];
#pragma unroll
  for (int d = 0; d < 8; ++d) xr[d] = xp[d];
  size_t base = (size_t)t * 8;  // t == (b*H + h)*S + s
#pragma unroll
  for (int e = 0; e < 8; ++e) {
    float aq = 0.f, ak = 0.f, av = 0.f;
#pragma unroll
    for (int d = 0; d < 8; ++d) {
      aq += xr[d] * wq[e * 8 + d];
      ak += xr[d] * wk[e * 8 + d];
      av += xr[d] * wv[e * 8 + d];
    }
    q[base + e] = aq;
    k[base + e] = ak;
    v[base + e] = av;
  }
}

// ---------------------------------------------------------------------------
// Flash attention, transposed formulation. 1 wave/block, 32 queries/block
// (two 16-query tiles), 128 key-tiles of 16.
//
//  S^T = K_tile x Q_tile^T : D(kk, m) -> lane(lo,hi) owns query m=lo,
//                            physical keys kk = 8*hi + r  (r = VGPR index).
//  Softmax per query: 7-op register tree + one shfl_xor(16) partner combine.
//  O^T = V^T x P^T with key permutation perm(4c+2hi+j) = 8hi+2c+j baked into
//  the V^T (A-operand) addressing, so the P^T B-operand for chunk c is just
//  the lane's own (p[2c], p[2c+1]). Zero LDS, zero barriers.
//  O^T D-layout leaves query lo's 8-dim output in lane lo (hi=0) -> 2x b128.
// ---------------------------------------------------------------------------
__global__ __launch_bounds__(32) void attn_kernel(const float* __restrict__ q,
                                                  const float* __restrict__ k,
                                                  const float* __restrict__ v,
                                                  float* __restrict__ o) {
  const int l = threadIdx.x;
  const int lo = l & 15;
  const int hi = l >> 4;
  const int bh = blockIdx.x >> 6;    // 64 q-blocks per (b,h)
  const int qblk = blockIdx.x & 63;  // 32 queries each
  const int b = bh >> 2;
  const int h = bh & 3;
  const float* qp = q + (size_t)bh * kS * kD;
  const float* kp = k + (size_t)bh * kS * kD;
  const float* vp = v + (size_t)bh * kS * kD;
  const float scale = 0.3535533905932738f;  // 1/sqrt(8)

  // Q tiles as B operand, pre-scaled by 1/sqrt(D):
  // lane(col m=lo, half hi) holds dims {2hi,2hi+1} and +4.
  v2f qb_[2][2];
#pragma unroll
  for (int t = 0; t < 2; ++t) {
    int qrow = qblk * 32 + t * 16 + lo;
    const float* qr = qp + (size_t)qrow * 8 + 2 * hi;
    qb_[t][0] = *(const v2f*)(qr)*scale;
    qb_[t][1] = *(const v2f*)(qr + 4) * scale;
  }

  v8f Oacc[2];
  float Mx[2], Lx[2];
#pragma unroll
  for (int t = 0; t < 2; ++t) {
#pragma unroll
    for (int r = 0; r < 8; ++r) Oacc[t][r] = 0.f;
    Mx[t] = -3.0e38f;
    Lx[t] = 0.f;
  }

  for (int kt = 0; kt < kS / 16; ++kt) {
    const int k0 = kt * 16;
    // K tile as A operand: lane(row kk=lo, half hi) holds dims {2hi,2hi+1},+4.
    const float* kr = kp + (size_t)(k0 + lo) * 8 + 2 * hi;
    v2f ka0 = *(const v2f*)(kr);
    v2f ka1 = *(const v2f*)(kr + 4);
    // V^T tile as A operand with key permutation:
    // chunk c, lane(row e=lo, half hi): rows k0 + 8*hi + 2c + {0,1}, col e=lo.
    // Rows e=8..15 are padding (zero).
    v2f va[4];
#pragma unroll
    for (int c = 0; c < 4; ++c) {
      int r0 = k0 + 8 * hi + 2 * c;
      va[c].x = (lo < 8) ? vp[(size_t)r0 * 8 + lo] : 0.f;
      va[c].y = (lo < 8) ? vp[(size_t)(r0 + 1) * 8 + lo] : 0.f;
    }

#pragma unroll
    for (int t = 0; t < 2; ++t) {
      // S^T(kk, m) = sum_d K[k0+kk][d] * (Q[q0+m][d] / sqrt(D))
      v8f st;
#pragma unroll
      for (int r = 0; r < 8; ++r) st[r] = 0.f;
      st = wmma_f32_16x16x4(ka0, qb_[t][0], st);
      st = wmma_f32_16x16x4(ka1, qb_[t][1], st);

      // lane-local softmax for query m = lo over this tile's 16 keys
      float lm = st[0];
#pragma unroll
      for (int r = 1; r < 8; ++r) lm = fmaxf(lm, st[r]);
      lm = fmaxf(lm, __shfl_xor(lm, 16));  // combine the two key-halves
      float nm = fmaxf(Mx[t], lm);
      float f = __expf(Mx[t] - nm);
      float p[8];
      float ls = 0.f;
#pragma unroll
      for (int r = 0; r < 8; ++r) {
        p[r] = __expf(st[r] - nm);
        ls += p[r];
      }
      ls += __shfl_xor(ls, 16);
      Lx[t] = Lx[t] * f + ls;
      Mx[t] = nm;
#pragma unroll
      for (int r = 0; r < 8; ++r) Oacc[t][r] *= f;

      // O^T += V^T x P^T ; B chunk c is the lane's own (p[2c], p[2c+1])
#pragma unroll
      for (int c = 0; c < 4; ++c) {
        v2f pb;
        pb.x = p[2 * c];
        pb.y = p[2 * c + 1];
        Oacc[t] = wmma_f32_16x16x4(va[c], pb, Oacc[t]);
      }
    }
  }

  // epilogue: lane lo (hi==0) owns O^T(e=0..7, m=lo) = full head output of
  // query q0+lo -> normalize and store as two b128.
  if (hi == 0) {
#pragma unroll
    for (int t = 0; t < 2; ++t) {
      int srow = qblk * 32 + t * 16 + lo;
      float inv = 1.f / Lx[t];
      float* op = o + ((size_t)b * kS + srow) * kE + h * kD;
      float4 o0 = {Oacc[t][0] * inv, Oacc[t][1] * inv, Oacc[t][2] * inv,
                   Oacc[t][3] * inv};
      float4 o1 = {Oacc[t][4] * inv, Oacc[t][5] * inv, Oacc[t][6] * inv,
                   Oacc[t][7] * inv};
      *(float4*)(op) = o0;
      *(float4*)(op + 4) = o1;
    }
  }
}

// ---------------------------------------------------------------------------
// n1 = LayerNorm(o @ Wo^T + bo + x) ; one thread per token
// ---------------------------------------------------------------------------
__global__ void proj_ln1_kernel(const float* __restrict__ o,
                                const float* __restrict__ x,
                                const float* __restrict__ wo,
                                const float* __restrict__ bo,
                                const float* __restrict__ g1,
                                const float* __restrict__ b1,
                                float* __restrict__ n1) {
  int tok = blockIdx.x * blockDim.x + threadIdx.x;  // B*S = 16384
  const float* orow = o + (size_t)tok * kE;
  const float* xrow = x + (size_t)tok * kE;
  float orr[32];
#pragma unroll
  for (int j = 0; j < 32; ++j) orr[j] = orow[j];
  float tv[32];
  float mean = 0.f;
#pragma unroll
  for (int e = 0; e < 32; ++e) {
    float acc = bo[e];
#pragma unroll
    for (int j = 0; j < 32; ++j) acc += orr[j] * wo[e * 32 + j];
    tv[e] = acc + xrow[e];
    mean += tv[e];
  }
  mean *= (1.f / 32.f);
  float var = 0.f;
#pragma unroll
  for (int e = 0; e < 32; ++e) {
    float d = tv[e] - mean;
    var += d * d;
  }
  var *= (1.f / 32.f);
  float inv = rsqrtf(var + kEps);
  float* out = n1 + (size_t)tok * kE;
#pragma unroll
  for (int e = 0; e < 32; ++e) out[e] = (tv[e] - mean) * inv * g1[e] + b1[e];
}

// ---------------------------------------------------------------------------
// ff = relu(n1 @ W1^T + bf1) @ W2^T + bf2 (scalar per token, broadcast over E)
// out = LayerNorm(ff + n1) ; one thread per token
// ---------------------------------------------------------------------------
__global__ void ffn_ln2_kernel(const float* __restrict__ n1,
                               const float* __restrict__ w1,
                               const float* __restrict__ bf1,
                               const float* __restrict__ w2,
                               const float* __restrict__ bf2,
                               const float* __restrict__ g2,
                               const float* __restrict__ b2,
                               float* __restrict__ out) {
  int tok = blockIdx.x * blockDim.x + threadIdx.x;  // B*S
  const float* nrow = n1 + (size_t)tok * kE;
  float nr[32];
#pragma unroll
  for (int e = 0; e < 32; ++e) nr[e] = nrow[e];
  float ff = bf2[0];
#pragma unroll 4
  for (int j = 0; j < kFF; ++j) {
    float acc = bf1[j];
#pragma unroll
    for (int e = 0; e < 32; ++e) acc += nr[e] * w1[j * 32 + e];
    acc = fmaxf(acc, 0.f);
    ff += acc * w2[j];
  }
  float mean = 0.f;
#pragma unroll
  for (int e = 0; e < 32; ++e) mean += ff + nr[e];
  mean *= (1.f / 32.f);
  float var = 0.f;
#pragma unroll
  for (int e = 0; e < 32; ++e) {
    float d = (ff + nr[e]) - mean;
    var += d * d;
  }
  var *= (1.f / 32.f);
  float inv = rsqrtf(var + kEps);
  float* op = out + (size_t)tok * kE;
#pragma unroll
  for (int e = 0; e < 32; ++e)
    op[e] = ((ff + nr[e]) - mean) * inv * g2[e] + b2[e];
}

// ---------------------------------------------------------------------------
extern "C" void kernel_launch(void* const* d_in, const int* in_sizes, int n_in,
                              void* d_out, int out_size, void* d_ws,
                              size_t ws_size, hipStream_t stream) {
  const float* src = (const float*)d_in[0];
  const float* Wq = (const float*)d_in[1];
  const float* Wk = (const float*)d_in[2];
  const float* Wv = (const float*)d_in[3];
  const float* Wo = (const float*)d_in[4];
  const float* bo = (const float*)d_in[5];
  const float* g1 = (const float*)d_in[6];
  const float* b1 = (const float*)d_in[7];
  const float* W1 = (const float*)d_in[8];
  const float* bf1 = (const float*)d_in[9];
  const float* W2 = (const float*)d_in[10];
  const float* bf2 = (const float*)d_in[11];
  const float* g2 = (const float*)d_in[12];
  const float* b2 = (const float*)d_in[13];
  float* out = (float*)d_out;
  float* ws = (float*)d_ws;

  const size_t NTOK = (size_t)kB * kS;  // 16384
  const size_t NX = NTOK * kE;          // 524288
  float* xbuf = ws;
  float* qb = ws + NX;
  float* kb = ws + 2 * NX;
  float* vb = ws + 3 * NX;
  float* ob = ws + 4 * NX;
  float* n1b = ws + 5 * NX;

  posadd_kernel<<<dim3(NX / 256), dim3(256), 0, stream>>>(src, xbuf);

  for (int l = 0; l < kL; ++l) {
    qkv_kernel<<<dim3((kB * kH * kS) / 256), dim3(256), 0, stream>>>(
        xbuf, Wq + l * kD * kD, Wk + l * kD * kD, Wv + l * kD * kD, qb, kb, vb);

    attn_kernel<<<dim3(kB * kH * (kS / 32)), dim3(32), 0, stream>>>(qb, kb, vb,
                                                                    ob);

    proj_ln1_kernel<<<dim3(NTOK / 256), dim3(256), 0, stream>>>(
        ob, xbuf, Wo + l * kE * kE, bo + l * kE, g1 + l * kE, b1 + l * kE, n1b);

    float* dst = (l == kL - 1) ? out : xbuf;
    ffn_ln2_kernel<<<dim3(NTOK / 256), dim3(256), 0, stream>>>(
        n1b, W1 + (size_t)l * kFF * kE, bf1 + l * kFF, W2 + l * kFF, bf2 + l,
        g2 + l * kE, b2 + l * kE, dst);
  }
}